// GCNConvDiagDGL_11682311045157
// MI455X (gfx1250) — compile-verified
//
#include <hip/hip_runtime.h>
#include <cstdint>
#include <cstddef>

// ---------------------------------------------------------------------------
// GCN diag-conv (copy_u · segment_sum) for MI455X (gfx1250, wave32).
//
// Pipeline (all on `stream`, workspace in d_ws):
//   1) k_zero_i32      : deg[] = 0
//   2) k_count         : deg[dst[e]]++            (int atomics, ~1.6M)
//   3) k_scan_blocks   : per-1024-tile exclusive scan of deg -> offs
//   4) k_scan_partials : scan the <=256 tile totals (single block)
//   5) k_finalize      : offs += tile base; cursor = offs
//   6) k_place         : esrc[atomicInc(cursor[dst[e]])] = src[e]  (CSR by dst)
//   7) k_aggregate     : wave-per-node pull reduction; feature rows gathered
//                        via a 4-deep async global->LDS ring (ASYNCcnt),
//                        one float2 store per lane per node. No f32 atomics.
// ---------------------------------------------------------------------------

#define THREADS       256
#define SCAN_ITEMS    4
#define SCAN_TILE     (THREADS * SCAN_ITEMS)   // 1024 elements per scan block
#define MAX_PARTIALS  256                      // supports N <= 262144 nodes
#define WPB           8                        // waves per block (wave32)
#define DEPTH         4                        // async gather pipeline depth

__global__ void k_zero_i32(int* __restrict__ p, int n) {
  int i = blockIdx.x * blockDim.x + threadIdx.x;
  if (i < n) p[i] = 0;
}

__global__ void k_count(const int* __restrict__ dst, int* __restrict__ deg, int E) {
  int e = blockIdx.x * blockDim.x + threadIdx.x;
  if (e < E) atomicAdd(&deg[dst[e]], 1);
}

// Exclusive scan of a 1024-element tile per block; tile totals -> partials.
__global__ void k_scan_blocks(const int* __restrict__ in, int* __restrict__ out,
                              int* __restrict__ partials, int n) {
  __shared__ int sums[THREADS];
  const int tid  = threadIdx.x;
  const int base = blockIdx.x * SCAN_TILE + tid * SCAN_ITEMS;

  int loc[SCAN_ITEMS];
  int s = 0;
#pragma unroll
  for (int k = 0; k < SCAN_ITEMS; ++k) {
    int idx = base + k;
    int v = (idx < n) ? in[idx] : 0;
    loc[k] = s;           // thread-local exclusive prefix
    s += v;
  }
  int x = s;              // becomes inclusive prefix over thread chunks
  sums[tid] = x;
  __syncthreads();
  for (int off = 1; off < THREADS; off <<= 1) {
    int t = (tid >= off) ? sums[tid - off] : 0;
    __syncthreads();
    x += t;
    sums[tid] = x;
    __syncthreads();
  }
  const int texcl = x - s;  // exclusive prefix of this thread's chunk
#pragma unroll
  for (int k = 0; k < SCAN_ITEMS; ++k) {
    int idx = base + k;
    if (idx < n) out[idx] = texcl + loc[k];
  }
  if (tid == THREADS - 1) partials[blockIdx.x] = x;  // tile total
}

// Exclusive scan (in place) of <=256 tile totals, one block of 256 threads.
__global__ void k_scan_partials(int* __restrict__ p, int nb) {
  __shared__ int sums[MAX_PARTIALS];
  const int tid = threadIdx.x;
  int v = (tid < nb) ? p[tid] : 0;
  int x = v;
  sums[tid] = x;
  __syncthreads();
  for (int off = 1; off < MAX_PARTIALS; off <<= 1) {
    int t = (tid >= off) ? sums[tid - off] : 0;
    __syncthreads();
    x += t;
    sums[tid] = x;
    __syncthreads();
  }
  if (tid < nb) p[tid] = x - v;
}

__global__ void k_finalize(int* __restrict__ offs, int* __restrict__ cursor,
                           const int* __restrict__ partials, int n) {
  int i = blockIdx.x * blockDim.x + threadIdx.x;
  if (i < n) {
    int o = offs[i] + partials[i / SCAN_TILE];
    offs[i]   = o;
    cursor[i] = o;
  }
}

__global__ void k_place(const int* __restrict__ src, const int* __restrict__ dst,
                        int* __restrict__ cursor, int* __restrict__ esrc, int E) {
  int e = blockIdx.x * blockDim.x + threadIdx.x;
  if (e < E) {
    int pos = atomicAdd(&cursor[dst[e]], 1);
    esrc[pos] = src[e];
  }
}

// Issue an async 8-byte global->LDS copy for this lane. The s_wait_dscnt 0
// guard orders prior DS reads of the ring slot against the async DMA write
// (ISA: async LDS writes are unordered vs DS ops). Tracked by ASYNCcnt.
__device__ __forceinline__ void async_gather_b64(const float* gaddr, void* lds) {
  unsigned l = (unsigned)(uint64_t)(uintptr_t)lds;   // low 32 bits = LDS offset
  uint64_t g = (uint64_t)(uintptr_t)gaddr;
  asm volatile("s_wait_dscnt 0x0\n\t"
               "global_load_async_to_lds_b64 %0, %1, off"
               :: "v"(l), "v"(g) : "memory");
}

// One wave per destination node. Lane L owns feature dims {2L, 2L+1}.
// Incoming source rows stream through a DEPTH-deep async LDS ring:
//   fill   : issue min(cnt, DEPTH) gathers
//   steady : s_wait_asynccnt DEPTH-1 (oldest landed) -> read -> reissue slot
//   drain  : s_wait_asynccnt 0, then wait-free reads of the tail
__global__ void k_aggregate(const float* __restrict__ feat,
                            const float* __restrict__ W,
                            const int* __restrict__ offs,
                            const int* __restrict__ deg,
                            const int* __restrict__ esrc,
                            float* __restrict__ out, int N) {
  __shared__ float2 stage[WPB][DEPTH][32];   // 8 KB: per-wave 4-slot ring
  const int lane = threadIdx.x & 31;
  const int wv   = threadIdx.x >> 5;
  const int gw   = blockIdx.x * WPB + wv;
  const int nw   = gridDim.x * WPB;
  const float2 w = ((const float2*)W)[lane];   // W is 64 floats: 2 per lane

  for (int node = gw; node < N; node += nw) {
    const int start = offs[node];          // uniform per wave -> scalar loads
    const int cnt   = deg[node];
    float a0 = 0.f, a1 = 0.f;
    if (cnt > 0) {
      const int P = cnt < DEPTH ? cnt : DEPTH;
      for (int j = 0; j < P; ++j)
        async_gather_b64(feat + (size_t)esrc[start + j] * 64u + 2u * lane,
                         &stage[wv][j][lane]);
      int i = 0;
      for (; i < cnt - DEPTH; ++i) {       // steady state: 4 gathers in flight
        asm volatile("s_wait_asynccnt 0x3" ::: "memory");   // oldest landed
        float2 v = stage[wv][i & (DEPTH - 1)][lane];
        a0 += v.x;
        a1 += v.y;
        async_gather_b64(feat + (size_t)esrc[start + i + DEPTH] * 64u + 2u * lane,
                         &stage[wv][i & (DEPTH - 1)][lane]);
      }
      asm volatile("s_wait_asynccnt 0x0" ::: "memory");     // drain pipeline
      for (; i < cnt; ++i) {
        float2 v = stage[wv][i & (DEPTH - 1)][lane];
        a0 += v.x;
        a1 += v.y;
      }
    }
    // Scale once after the sum:  sum(h_src)·W  ==  sum(h_src·W) for diag W.
    ((float2*)out)[(size_t)node * 32 + lane] = make_float2(a0 * w.x, a1 * w.y);
  }
}

extern "C" void kernel_launch(void* const* d_in, const int* in_sizes, int n_in,
                              void* d_out, int out_size, void* d_ws, size_t ws_size,
                              hipStream_t stream) {
  (void)n_in; (void)out_size; (void)ws_size;
  const float* feat = (const float*)d_in[0];
  const float* W    = (const float*)d_in[1];
  const int*   src  = (const int*)d_in[2];
  const int*   dst  = (const int*)d_in[3];
  float*       out  = (float*)d_out;

  const int D = in_sizes[1];           // 64
  const int N = in_sizes[0] / D;       // 100000
  const int E = in_sizes[2];           // 1600000

  // Workspace layout (all int32): deg | offs | cursor | partials | esrc
  // total = 3N + 256 + E ints ~= 7.6 MB.
  int* deg      = (int*)d_ws;
  int* offs     = deg + N;
  int* cursor   = offs + N;
  int* partials = cursor + N;
  int* esrc     = partials + MAX_PARTIALS;

  const int nb = (N + SCAN_TILE - 1) / SCAN_TILE;   // <= 256 tiles

  k_zero_i32     <<<(N + THREADS - 1) / THREADS, THREADS, 0, stream>>>(deg, N);
  k_count        <<<(E + THREADS - 1) / THREADS, THREADS, 0, stream>>>(dst, deg, E);
  k_scan_blocks  <<<nb, THREADS, 0, stream>>>(deg, offs, partials, N);
  k_scan_partials<<<1, MAX_PARTIALS, 0, stream>>>(partials, nb);
  k_finalize     <<<(N + THREADS - 1) / THREADS, THREADS, 0, stream>>>(offs, cursor, partials, N);
  k_place        <<<(E + THREADS - 1) / THREADS, THREADS, 0, stream>>>(src, dst, cursor, esrc, E);
  k_aggregate    <<<2048, THREADS, 0, stream>>>(feat, W, offs, deg, esrc, out, N);
}